// CHN_MU_MISO_84636625535007
// MI455X (gfx1250) — compile-verified
//
#include <hip/hip_runtime.h>

typedef __attribute__((ext_vector_type(2))) float v2f;
typedef __attribute__((ext_vector_type(4))) float v4f;
typedef __attribute__((ext_vector_type(8))) float v8f;

#define U_ 4
#define NT_ 8

// Fold W,H,P,stddev into 20 coefficients:
//   coef[u*4+v] = A[u][v] = (sum_n H[n,u]*W[n,v]*sqrtP[v]) / amp[u]
//   coef[16+u]  = stddev[u] / amp[u]
__global__ void chn_prep_kernel(const float* __restrict__ W, const float* __restrict__ H,
                                const float* __restrict__ P, const float* __restrict__ stddev,
                                float* __restrict__ coef)
{
    if (threadIdx.x == 0 && blockIdx.x == 0) {
        float sqrtP[U_], amp[U_];
        for (int u = 0; u < U_; ++u) sqrtP[u] = sqrtf(P[u]);
        for (int u = 0; u < U_; ++u) {
            float d = 0.0f;
            for (int n = 0; n < NT_; ++n) d += H[n * U_ + u] * W[n * U_ + u];
            amp[u] = sqrtP[u] * d;
        }
        for (int u = 0; u < U_; ++u) {
            for (int v = 0; v < U_; ++v) {
                float mm = 0.0f;
                for (int n = 0; n < NT_; ++n) mm += H[n * U_ + u] * W[n * U_ + v];
                coef[u * U_ + v] = mm * sqrtP[v] / amp[u];
            }
            coef[16 + u] = stddev[u] / amp[u];
        }
    }
}

// Streaming kernel: out[u, e] = sum_v A[u][v] * x[v, e] + s[u]*noise[u, e]
// done as D = A(16x4, rows 4..15 zero) * B(4x16) + C via V_WMMA_F32_16X16X4_F32.
// Each wave iteration covers 64 consecutive element-columns (4 WMMAs, one per
// float4 component), with all global traffic in 256B-contiguous segments.
__global__ void __launch_bounds__(256) chn_stream_kernel(
    const float* __restrict__ x, const float* __restrict__ noise,
    const float* __restrict__ coef, float* __restrict__ out,
    const long long planeElems, const long long nTiles)
{
    const int lane = threadIdx.x & 31;
    const int m    = lane & 15;          // matrix row index within half-wave
    const int kh   = (lane >> 4) << 1;   // K pair held by this lane: 0/1 or 2/3

    const long long waveId     = (long long)blockIdx.x * (blockDim.x >> 5) + (threadIdx.x >> 5);
    const long long totalWaves = (long long)gridDim.x * (blockDim.x >> 5);

    // A operand: lane (half, m) holds A[M=m][K=kh], A[M=m][K=kh+1]; rows >=4 are zero.
    v2f a; a.x = 0.0f; a.y = 0.0f;
    if (m < U_) {
        a.x = coef[m * U_ + kh];
        a.y = coef[m * U_ + kh + 1];
    }
    const float s0 = coef[16], s1 = coef[17], s2 = coef[18], s3 = coef[19];

    const float* xa_p = x + (long long)kh * planeElems;        // user kh
    const float* xb_p = x + (long long)(kh + 1) * planeElems;  // user kh+1

    for (long long t = waveId; t < nTiles; t += totalWaves) {
        const long long off = t * 64 + 4 * m;   // element-column base for this lane

        // B source: users (kh, kh+1), 4 consecutive columns per lane (float4).
        const v4f xa = *(const v4f*)(xa_p + off);
        const v4f xb = *(const v4f*)(xb_p + off);

        // C source: rows 0..3 live on lanes 0..15; lanes 16..31 keep zeros so
        // C rows 8..15 are zero without predicating the WMMA.
        v4f n0 = {0.f,0.f,0.f,0.f}, n1 = n0, n2 = n0, n3 = n0;
        if (lane < 16) {
            n0 = *(const v4f*)(noise + off);
            n1 = *(const v4f*)(noise + planeElems + off);
            n2 = *(const v4f*)(noise + 2 * planeElems + off);
            n3 = *(const v4f*)(noise + 3 * planeElems + off);
        }

        v4f o0, o1, o2, o3;
        #pragma unroll
        for (int j = 0; j < 4; ++j) {
            v2f b; b.x = xa[j]; b.y = xb[j];     // B[K=kh][N=m], B[K=kh+1][N=m], column = elem off+j
            v8f c;
            c[0] = s0 * n0[j];
            c[1] = s1 * n1[j];
            c[2] = s2 * n2[j];
            c[3] = s3 * n3[j];
            c[4] = 0.0f; c[5] = 0.0f; c[6] = 0.0f; c[7] = 0.0f;
            v8f d = __builtin_amdgcn_wmma_f32_16x16x4_f32(
                false, a, false, b, (short)0, c, false, false);
            o0[j] = d[0]; o1[j] = d[1]; o2[j] = d[2]; o3[j] = d[3];  // rows 0..3 (lanes 0..15)
        }

        if (lane < 16) {
            *(v4f*)(out + off)                  = o0;
            *(v4f*)(out + planeElems + off)     = o1;
            *(v4f*)(out + 2 * planeElems + off) = o2;
            *(v4f*)(out + 3 * planeElems + off) = o3;
        }
    }
}

extern "C" void kernel_launch(void* const* d_in, const int* in_sizes, int n_in,
                              void* d_out, int out_size, void* d_ws, size_t ws_size,
                              hipStream_t stream)
{
    const float* x      = (const float*)d_in[0];  // [U, B, C] f32
    const float* W      = (const float*)d_in[1];  // [NT, U]
    const float* H      = (const float*)d_in[2];  // [NT, U]
    const float* P      = (const float*)d_in[3];  // [U]
    const float* stddev = (const float*)d_in[4];  // [U]
    const float* noise  = (const float*)d_in[5];  // [U, B, C]
    float* out  = (float*)d_out;
    float* coef = (float*)d_ws;                   // 20 floats

    const long long planeElems = (long long)in_sizes[0] / U_;  // B*C = 6,291,456
    const long long nTiles     = planeElems / 64;              // exact: 98,304

    chn_prep_kernel<<<1, 32, 0, stream>>>(W, H, P, stddev, coef);
    // 1536 blocks * 8 waves = 12288 waves; 8 tiles per wave amortizes setup.
    chn_stream_kernel<<<1536, 256, 0, stream>>>(x, noise, coef, out, planeElems, nTiles);
}